// SparseProjector_21036749816194
// MI455X (gfx1250) — compile-verified
//
#include <hip/hip_runtime.h>
#include <stdint.h>

#ifndef __has_builtin
#define __has_builtin(x) 0
#endif

// Problem constants (match reference)
#define B_      8
#define D_      65536
#define F_      16
#define M_      4096

// Tuning
#define CHUNKS  16        // nonzero chunks per batch -> 16*8 = 128 workgroups, 1/WGP (LDS-bound)
#define BLOCK   1024      // 32 wave32 waves per WGP
#define STAGE   1024      // nonzeros per double-buffer stage

// LDS layout: padded accumulator + double-buffered index staging
#define ACC_STRIDE 17                          // 16->17 dword row stride: 17 coprime 64 banks
#define ACC_DWORDS (M_ * ACC_STRIDE)           // 69632 dwords = 278528 B
#define LDS_BYTES  (ACC_DWORDS * 4 + 2 * STAGE * 3 * 4)  // 303104 B <= 320KB WGP LDS

#if __has_builtin(__builtin_amdgcn_global_load_async_to_lds_b32) && \
    __has_builtin(__builtin_amdgcn_s_wait_asynccnt)
#define USE_ASYNC_LDS 1
#else
#define USE_ASYNC_LDS 0
#endif

#if USE_ASYNC_LDS
#define WAIT_ASYNC(n) __builtin_amdgcn_s_wait_asynccnt(n)
typedef __attribute__((address_space(1))) int* gas_ptr_t;   // global AS int*
typedef __attribute__((address_space(3))) int* las_ptr_t;   // LDS AS int*
__device__ __forceinline__ void async_b32(const void* g, void* l) {
  __builtin_amdgcn_global_load_async_to_lds_b32((gas_ptr_t)g, (las_ptr_t)l, 0, 0);
}
#else
#define WAIT_ASYNC(n) ((void)0)
__device__ __forceinline__ void async_b32(const void* g, void* l) {
  *(uint32_t*)l = *(const uint32_t*)g;   // sync fallback (still a prefetch pattern)
}
#endif

__device__ __forceinline__ int imin(int a, int b) { return a < b ? a : b; }

// Main kernel first in the module so the disasm snippet shows its prologue
// (async staging copies should appear as global_load_async_to_lds_b32).
__global__ void __launch_bounds__(BLOCK)
sparse_projector_kernel(const float* __restrict__ x,
                        const int*   __restrict__ rows,
                        const int*   __restrict__ cols,
                        const float* __restrict__ vals,
                        float*       __restrict__ out,
                        int nnz, int chunk)
{
  extern __shared__ char smem[];
  float* acc    = reinterpret_cast<float*>(smem);                 // 4096 x 17 f32
  int*   s_rows = reinterpret_cast<int*>(smem + ACC_DWORDS * 4);  // 2 x STAGE
  int*   s_cols = s_rows + 2 * STAGE;
  float* s_vals = reinterpret_cast<float*>(s_cols + 2 * STAGE);

  const int tid = threadIdx.x;
  const int b   = blockIdx.y;
  const int c0  = blockIdx.x * chunk;
  const int c1  = imin(c0 + chunk, nnz);
  const int nstage = (c1 - c0 + STAGE - 1) / STAGE;
  const float* xb = x + (size_t)b * (D_ * F_);

  // Prologue: async-stage index streams for stage 0 into buffer 0.
  // Indices are clamped (not predicated) so every wave issues exactly 3 async
  // ops per stage -> per-wave ASYNCcnt bookkeeping below is exact.
  {
    int g = imin(c0 + tid, nnz - 1);
    async_b32(rows + g, s_rows + tid);
    async_b32(cols + g, s_cols + tid);
    async_b32(vals + g, s_vals + tid);
  }

  // Zero the privatized per-batch accumulator (whole out[b] lives in LDS)
  // while the stage-0 index DMA is in flight.
  #pragma unroll
  for (int k = 0; k < ACC_DWORDS / BLOCK; ++k)
    acc[tid + k * BLOCK] = 0.f;

  for (int s = 0; s < nstage; ++s) {
    const int p = s & 1;
    if (s + 1 < nstage) {
      const int q = p ^ 1;
      int g = imin(c0 + (s + 1) * STAGE + tid, nnz - 1);
      async_b32(rows + g, s_rows + q * STAGE + tid);
      async_b32(cols + g, s_cols + q * STAGE + tid);
      async_b32(vals + g, s_vals + q * STAGE + tid);
      WAIT_ASYNC(3);   // 3 newer ops in flight; stage-s copies complete
    } else {
      WAIT_ASYNC(0);
    }
    __syncthreads();   // all waves' stage-s copies now visible in LDS

    const int cnt = c1 - (c0 + s * STAGE);
    if (tid < cnt) {
      const int   r = s_rows[p * STAGE + tid];
      const int   c = s_cols[p * STAGE + tid];
      const float v = s_vals[p * STAGE + tid];
      // Random 64B row gather (L2-resident: x is 32MB total)
      const float4* xr = reinterpret_cast<const float4*>(xb + (size_t)c * F_);
      float4 a0 = xr[0], a1 = xr[1], a2 = xr[2], a3 = xr[3];
      float* dst = acc + r * ACC_STRIDE;   // padded stride -> banks spread
      atomicAdd(dst + 0,  v * a0.x);  atomicAdd(dst + 1,  v * a0.y);
      atomicAdd(dst + 2,  v * a0.z);  atomicAdd(dst + 3,  v * a0.w);
      atomicAdd(dst + 4,  v * a1.x);  atomicAdd(dst + 5,  v * a1.y);
      atomicAdd(dst + 6,  v * a1.z);  atomicAdd(dst + 7,  v * a1.w);
      atomicAdd(dst + 8,  v * a2.x);  atomicAdd(dst + 9,  v * a2.y);
      atomicAdd(dst + 10, v * a2.z);  atomicAdd(dst + 11, v * a2.w);
      atomicAdd(dst + 12, v * a3.x);  atomicAdd(dst + 13, v * a3.y);
      atomicAdd(dst + 14, v * a3.z);  atomicAdd(dst + 15, v * a3.w);
    }
    __syncthreads();   // buffer p free for reuse two stages later
  }

  // Flush privatized accumulator to global (16 chunk-WGs per batch -> atomics)
  float* outb = out + (size_t)b * (M_ * F_);
  #pragma unroll
  for (int k = 0; k < (M_ * F_) / BLOCK; ++k) {
    int idx = tid + k * BLOCK;
    int r = idx >> 4, f = idx & 15;
    atomicAdd(outb + idx, acc[r * ACC_STRIDE + f]);
  }
}

__global__ void sp_zero_out(float4* out4, int n4) {
  int i = blockIdx.x * blockDim.x + threadIdx.x;
  if (i < n4) out4[i] = make_float4(0.f, 0.f, 0.f, 0.f);
}

extern "C" void kernel_launch(void* const* d_in, const int* in_sizes, int n_in,
                              void* d_out, int out_size, void* d_ws, size_t ws_size,
                              hipStream_t stream) {
  const float* x    = (const float*)d_in[0];
  const int*   rows = (const int*)d_in[1];
  const int*   cols = (const int*)d_in[2];
  const float* vals = (const float*)d_in[3];
  float* out = (float*)d_out;
  const int nnz = in_sizes[1];

  // Allow >64KB dynamic LDS (idempotent, capture-safe host call)
  (void)hipFuncSetAttribute(reinterpret_cast<const void*>(sparse_projector_kernel),
                            hipFuncAttributeMaxDynamicSharedMemorySize, LDS_BYTES);

  // out must be zeroed every call (harness poisons d_out; flush is accumulate)
  int n4 = out_size / 4;
  sp_zero_out<<<(n4 + 255) / 256, 256, 0, stream>>>((float4*)out, n4);

  int chunk = (nnz + CHUNKS - 1) / CHUNKS;
  dim3 grid(CHUNKS, B_);
  sparse_projector_kernel<<<grid, dim3(BLOCK), LDS_BYTES, stream>>>(
      x, rows, cols, vals, out, nnz, chunk);
}